// ASAP_18657337933851
// MI455X (gfx1250) — compile-verified
//
#include <hip/hip_runtime.h>

typedef __attribute__((ext_vector_type(2))) float v2f;
typedef __attribute__((ext_vector_type(8))) float v8f;

#define BGRAPH 32
#define NGN    256
#define NN     8192          // BGRAPH*NGN
#define ETOT   131072        // edges
#define E2T    (ETOT + NN)   // + self loops
#define HF     128
#define KC1    205
#define KP1    208
#define KC2    164
#define KP2    176
#define NSLOPE 0.2f
#define TB     256

// ---------------- helpers ----------------
__device__ __forceinline__ float leaky(float x){ return x > 0.f ? x : NSLOPE * x; }
__device__ __forceinline__ unsigned fkey(float f){
  unsigned u = __float_as_uint(f);
  return (u & 0x80000000u) ? ~u : (u | 0x80000000u);
}
__device__ __forceinline__ float funkey(unsigned k){
  if (k == 0u) return -1e30f;
  unsigned u = (k & 0x80000000u) ? (k & 0x7fffffffu) : ~k;
  return __uint_as_float(u);
}

// ---------------- elementwise / scatter kernels ----------------
__global__ void k_fill(float* p, float v, long n){
  long i = (long)blockIdx.x * blockDim.x + threadIdx.x;
  if (i < n) p[i] = v;
}
__global__ void k_fillu(unsigned* p, unsigned v, long n){
  long i = (long)blockIdx.x * blockDim.x + threadIdx.x;
  if (i < n) p[i] = v;
}
__global__ void k_addc(const float* a, float c, float* o, int n){
  int i = blockIdx.x * blockDim.x + threadIdx.x;
  if (i < n) o[i] = a[i] + c;
}
__global__ void k_deg(const int* dst, float* deg, int E, int nloops){
  int i = blockIdx.x * blockDim.x + threadIdx.x;
  if (i >= E + nloops) return;
  int d = (i < E) ? dst[i] : (i - E);
  atomicAdd(&deg[d], 1.0f);
}
__global__ void k_scatter_add(const float* x, const int* src, const int* dst,
                              const float* w, float* out, int E, int nloops, int F){
  long i = (long)blockIdx.x * blockDim.x + threadIdx.x;
  long tot = (long)(E + nloops) * F;
  if (i >= tot) return;
  int e = (int)(i / F), f = (int)(i % F);
  int s = (e < E) ? src[e] : (e - E);
  int d = (e < E) ? dst[e] : (e - E);
  float v = x[(long)s * F + f];
  if (w) v *= w[e];
  atomicAdd(&out[(long)d * F + f], v);
}
__global__ void k_scatter_max(const float* x, const int* src, const int* dst,
                              unsigned* keys, int E, int nloops, int F){
  long i = (long)blockIdx.x * blockDim.x + threadIdx.x;
  long tot = (long)(E + nloops) * F;
  if (i >= tot) return;
  int e = (int)(i / F), f = (int)(i % F);
  int s = (e < E) ? src[e] : (e - E);
  int d = (e < E) ? dst[e] : (e - E);
  atomicMax(&keys[(long)d * F + f], fkey(x[(long)s * F + f]));
}
__global__ void k_decode(const unsigned* k, float* o, long n){
  long i = (long)blockIdx.x * blockDim.x + threadIdx.x;
  if (i >= n) return;
  unsigned kk = k[i];
  o[i] = funkey(kk);
}
__global__ void k_scale_rows(float* x, const float* deg, long n, int F){
  long i = (long)blockIdx.x * blockDim.x + threadIdx.x;
  if (i >= n) return;
  float d = deg[i / F];
  if (d < 1.f) d = 1.f;
  x[i] /= d;
}
__global__ void k_matvec(const float* X, int ld, const float* w, float* o, int rows, int K){
  int r = blockIdx.x * blockDim.x + threadIdx.x;
  if (r >= rows) return;
  const float* xp = X + (long)r * ld;
  float s = 0.f;
  for (int k = 0; k < K; ++k) s += xp[k] * w[k];
  o[r] = s;
}

// ---------------- edge softmax (sparse ASAP) ----------------
__global__ void k_edge_e(const int* src, const int* dst, const float* aq, const float* aj,
                         const float* batt, float* e, int E, int nloops){
  int i = blockIdx.x * blockDim.x + threadIdx.x;
  if (i >= E + nloops) return;
  int s = (i < E) ? src[i] : (i - E);
  int d = (i < E) ? dst[i] : (i - E);
  e[i] = leaky(aq[d] + aj[s] + *batt);
}
__global__ void k_edge_smax(const float* e, const int* dst, unsigned* keys, int E, int nloops){
  int i = blockIdx.x * blockDim.x + threadIdx.x;
  if (i >= E + nloops) return;
  int d = (i < E) ? dst[i] : (i - E);
  atomicMax(&keys[d], fkey(e[i]));
}
__global__ void k_edge_exp(float* e, const int* dst, const float* m, float* z, int E, int nloops){
  int i = blockIdx.x * blockDim.x + threadIdx.x;
  if (i >= E + nloops) return;
  int d = (i < E) ? dst[i] : (i - E);
  float ex = expf(e[i] - m[d]);
  e[i] = ex;
  atomicAdd(&z[d], ex);
}
__global__ void k_edge_norm(float* e, const int* dst, const float* z, int E, int nloops){
  int i = blockIdx.x * blockDim.x + threadIdx.x;
  if (i >= E + nloops) return;
  int d = (i < E) ? dst[i] : (i - E);
  e[i] /= z[d];
}
__global__ void k_edge_sub(const float* t3, const int* src, const int* dst, float* fit, int E, int nloops){
  int i = blockIdx.x * blockDim.x + threadIdx.x;
  if (i >= E + nloops) return;
  int s = (i < E) ? src[i] : (i - E);
  int d = (i < E) ? dst[i] : (i - E);
  atomicAdd(&fit[d], -t3[s]);
}
__global__ void k_fit_init(const float* t1, const float* t2, const float* deg,
                           const float* leb1, float* fit, int n){
  int i = blockIdx.x * blockDim.x + threadIdx.x;
  if (i < n) fit[i] = t1[i] + *leb1 + deg[i] * t2[i];
}
__global__ void k_sigmoid(float* f, int n){
  int i = blockIdx.x * blockDim.x + threadIdx.x;
  if (i < n) f[i] = 1.f / (1.f + expf(-f[i]));
}

// ---------------- top-k via rank selection ----------------
__global__ void k_topk(const float* fit, int* idx, int NGsize, int Ksel){
  __shared__ float sf[256];
  int b = blockIdx.x, i = threadIdx.x;
  sf[i] = fit[(long)b * NGsize + i];
  __syncthreads();
  float f = sf[i];
  int rank = 0;
  for (int j = 0; j < NGsize; ++j){
    float fj = sf[j];
    rank += (fj > f) || (fj == f && j < i);
  }
  if (rank < Ksel) idx[b * Ksel + rank] = i;
}
__global__ void k_gather_x(const float* xc, const float* fit, const int* idx,
                           float* xn, int NGin, int Ksel, int Kpad, int F){
  long i = (long)blockIdx.x * blockDim.x + threadIdx.x;
  long tot = (long)BGRAPH * Kpad * F;
  if (i >= tot) return;
  int f = (int)(i % F);
  long t = i / F;
  int r = (int)(t % Kpad);
  int b = (int)(t / Kpad);
  float v = 0.f;
  if (r < Ksel){
    int l = idx[b * Ksel + r];
    v = xc[((long)b * NGin + l) * F + f] * fit[(long)b * NGin + l];
  }
  xn[i] = v;
}
__global__ void k_build_SA(const int* src, const int* dst, const float* sc,
                           float* S, float* A, int E, int nloops){
  int i = blockIdx.x * blockDim.x + threadIdx.x;
  if (i >= E + nloops) return;
  int s = (i < E) ? src[i] : (i - E);
  int d = (i < E) ? dst[i] : (i - E);
  int g = s / NGN, sl = s % NGN, dl = d % NGN;
  long base = ((long)g * NGN + sl) * NGN + dl;
  atomicAdd(&S[base], sc[i]);
  atomicAdd(&A[base], 1.0f);
}
__global__ void k_gather_Sp(const float* S, const int* idx, float* Sp,
                            int rows, int cols, int Ksel, int Kpad){
  long i = (long)blockIdx.x * blockDim.x + threadIdx.x;
  long tot = (long)BGRAPH * rows * Kpad;
  if (i >= tot) return;
  int c = (int)(i % Kpad);
  long t = i / Kpad;
  int s = (int)(t % rows);
  int b = (int)(t / rows);
  float v = 0.f;
  if (c < Ksel) v = S[((long)b * rows + s) * cols + idx[b * Ksel + c]];
  Sp[i] = v;
}
__global__ void k_set_diag(float* A, int Kpad, int Ksel){
  int i = blockIdx.x * blockDim.x + threadIdx.x;
  if (i >= BGRAPH * Ksel) return;
  int b = i / Ksel, r = i % Ksel;
  A[((long)b * Kpad + r) * Kpad + r] = 1.0f;
}

// ---------------- dense-stage helpers ----------------
__global__ void k_dense_deg(const float* A, float* deg, int Kpad){
  int i = blockIdx.x * blockDim.x + threadIdx.x;
  if (i >= BGRAPH * Kpad) return;
  int b = i / Kpad, t = i % Kpad;
  const float* Ab = A + (long)b * Kpad * Kpad;
  float c = 0.f;
  for (int s = 0; s < Kpad; ++s) c += (Ab[(long)s * Kpad + t] != 0.f) ? 1.f : 0.f;
  deg[i] = c;
}
__global__ void k_dense_xq(const float* A, const float* x, float* xq, int Kpad, int F){
  long i = (long)blockIdx.x * blockDim.x + threadIdx.x;
  long tot = (long)BGRAPH * Kpad * F;
  if (i >= tot) return;
  int f = (int)(i % F);
  long t2 = i / F;
  int t = (int)(t2 % Kpad);
  int b = (int)(t2 / Kpad);
  const float* Ab = A + (long)b * Kpad * Kpad;
  const float* xb = x + (long)b * Kpad * F;
  float m = -1e30f;
  for (int s = 0; s < Kpad; ++s)
    if (Ab[(long)s * Kpad + t] != 0.f){ float v = xb[(long)s * F + f]; m = v > m ? v : m; }
  xq[i] = m;
}
__global__ void k_dense_softmax(const float* A, const float* aq, const float* aj,
                                const float* batt, float* sc, int Kpad){
  int i = blockIdx.x * blockDim.x + threadIdx.x;
  if (i >= BGRAPH * Kpad) return;
  int b = i / Kpad, t = i % Kpad;
  const float* Ab  = A  + (long)b * Kpad * Kpad;
  const float* ajb = aj + (long)b * Kpad;
  float qa = aq[i], ba = *batt;
  float m = -1e30f; int any = 0;
  for (int s = 0; s < Kpad; ++s)
    if (Ab[(long)s * Kpad + t] != 0.f){ float e = leaky(ajb[s] + qa + ba); if (e > m) m = e; any = 1; }
  float z = 0.f;
  if (any)
    for (int s = 0; s < Kpad; ++s)
      if (Ab[(long)s * Kpad + t] != 0.f) z += expf(leaky(ajb[s] + qa + ba) - m);
  float* scb = sc + (long)b * Kpad * Kpad;
  for (int s = 0; s < Kpad; ++s){
    float v = 0.f;
    if (any && Ab[(long)s * Kpad + t] != 0.f) v = expf(leaky(ajb[s] + qa + ba) - m) / z;
    scb[(long)s * Kpad + t] = v;
  }
}
__global__ void k_dense_fit(const float* t1, const float* t2, const float* t3,
                            const float* A, const float* deg, const float* leb1,
                            float* fit, int Kpad, int kvalid){
  int i = blockIdx.x * blockDim.x + threadIdx.x;
  if (i >= BGRAPH * Kpad) return;
  int b = i / Kpad, t = i % Kpad;
  if (t >= kvalid){ fit[i] = -1e30f; return; }
  const float* Ab  = A  + (long)b * Kpad * Kpad;
  const float* t3b = t3 + (long)b * Kpad;
  float ssum = 0.f;
  for (int s = 0; s < Kpad; ++s)
    if (Ab[(long)s * Kpad + t] != 0.f) ssum += t3b[s];
  float v = t1[i] + *leb1 + deg[i] * t2[i] - ssum;
  fit[i] = 1.f / (1.f + expf(-v));
}
__global__ void k_graph_sum(const float* h, float* z, int rpg, int valid, int F, int zld, int zoff){
  int b = blockIdx.x, f = threadIdx.x;
  if (f >= F) return;
  const float* hb = h + (long)b * rpg * F;
  float s = 0.f;
  for (int r = 0; r < valid; ++r) s += hb[(long)r * F + f];
  z[(long)b * zld + zoff + f] = s;
}

// ---------------- WMMA fp32 GEMMs (V_WMMA_F32_16X16X4_F32) ----------------
// one wave per 16x16 tile; C frag layout: VGPR r -> row (r + (lane>>4)*8), col (lane&15)
__global__ __launch_bounds__(32)
void k_wmma_gemm(const float* A1, int lda1, const float* W1,
                 const float* A2, int lda2, const float* W2,
                 const float* bias, float* C, int ldc,
                 int Ntot, int K, int relu){
  int tm = blockIdx.x, tn = blockIdx.y;
  int lane = threadIdx.x;
  int m  = lane & 15;
  int kk = (lane >> 4) << 1;
  v8f acc = {};
  {
    const float* Ap = A1 + (long)(tm * 16 + m) * lda1;
    const float* Wp = W1 + tn * 16 + m;
    __builtin_prefetch(Ap, 0, 1);
    __builtin_prefetch(Wp, 0, 1);
    for (int k0 = 0; k0 < K; k0 += 4){
      v2f a, b;
      a.x = Ap[k0 + kk];
      a.y = Ap[k0 + kk + 1];
      b.x = Wp[(long)(k0 + kk) * Ntot];
      b.y = Wp[(long)(k0 + kk + 1) * Ntot];
      acc = __builtin_amdgcn_wmma_f32_16x16x4_f32(false, a, false, b, (short)0, acc, false, false);
    }
  }
  if (A2){
    const float* Ap = A2 + (long)(tm * 16 + m) * lda2;
    const float* Wp = W2 + tn * 16 + m;
    for (int k0 = 0; k0 < K; k0 += 4){
      v2f a, b;
      a.x = Ap[k0 + kk];
      a.y = Ap[k0 + kk + 1];
      b.x = Wp[(long)(k0 + kk) * Ntot];
      b.y = Wp[(long)(k0 + kk + 1) * Ntot];
      acc = __builtin_amdgcn_wmma_f32_16x16x4_f32(false, a, false, b, (short)0, acc, false, false);
    }
  }
  int row0 = tm * 16 + ((lane >> 4) << 3);
  int col  = tn * 16 + m;
  float bv = bias ? bias[col] : 0.f;
  for (int r = 0; r < 8; ++r){
    float v = acc[r] + bv;
    if (relu) v = fmaxf(v, 0.f);
    C[(long)(row0 + r) * ldc + col] = v;
  }
}

// batched per-graph GEMM, optional transpose on A: C[b] = opA(A[b]) @ B[b]
__global__ __launch_bounds__(32)
void k_wmma_bgemm(const float* A, long sA, int lda, int transA,
                  const float* Bm, long sB, int ldb,
                  float* C, long sC, int ldc, int K){
  int tm = blockIdx.x, tn = blockIdx.y, bb = blockIdx.z;
  const float* Ab = A  + (long)bb * sA;
  const float* Bb = Bm + (long)bb * sB;
  float*       Cb = C  + (long)bb * sC;
  int lane = threadIdx.x;
  int m  = lane & 15;
  int kk = (lane >> 4) << 1;
  v8f acc = {};
  __builtin_prefetch(Bb + tn * 16 + m, 0, 1);
  for (int k0 = 0; k0 < K; k0 += 4){
    v2f a, b;
    if (transA){
      a.x = Ab[(long)(k0 + kk) * lda + tm * 16 + m];
      a.y = Ab[(long)(k0 + kk + 1) * lda + tm * 16 + m];
    } else {
      a.x = Ab[(long)(tm * 16 + m) * lda + k0 + kk];
      a.y = Ab[(long)(tm * 16 + m) * lda + k0 + kk + 1];
    }
    b.x = Bb[(long)(k0 + kk) * ldb + tn * 16 + m];
    b.y = Bb[(long)(k0 + kk + 1) * ldb + tn * 16 + m];
    acc = __builtin_amdgcn_wmma_f32_16x16x4_f32(false, a, false, b, (short)0, acc, false, false);
  }
  int row0 = tm * 16 + ((lane >> 4) << 3);
  int col  = tn * 16 + m;
  for (int r = 0; r < 8; ++r)
    Cb[(long)(row0 + r) * ldc + col] = acc[r];
}

// ---------------- head: 64->2 + log_softmax ----------------
__global__ void k_head(const float* z2, const float* w3, const float* b3, float* out){
  int b = threadIdx.x;
  if (b >= BGRAPH) return;
  const float* zp = z2 + b * 64;
  float l0 = b3[0], l1 = b3[1];
  for (int k = 0; k < 64; ++k){ l0 += zp[k] * w3[k * 2]; l1 += zp[k] * w3[k * 2 + 1]; }
  float m = fmaxf(l0, l1);
  float lse = m + logf(expf(l0 - m) + expf(l1 - m));
  out[b * 2]     = l0 - lse;
  out[b * 2 + 1] = l1 - lse;
}

// ---------------- host ----------------
static inline unsigned int gdl(long long n, int b){ return (unsigned int)((n + b - 1) / b); }

extern "C" void kernel_launch(void* const* d_in, const int* in_sizes, int n_in,
                              void* d_out, int out_size, void* d_ws, size_t ws_size,
                              hipStream_t stream){
  (void)in_sizes; (void)out_size; (void)ws_size;
  // inputs: x, then params flattened (jax pytree: sorted dict keys), then src, dst
  const float* x = (const float*)d_in[0];
  int pi = 1;
  const float* b1 = (const float*)d_in[pi++];
  const float* b2 = (const float*)d_in[pi++];
  const float* b3 = (const float*)d_in[pi++];
  const float* c1_brel  = (const float*)d_in[pi++];
  const float* c1_wrel  = (const float*)d_in[pi++];
  const float* c1_wroot = (const float*)d_in[pi++];
  const float *cv_brel[4], *cv_wrel[4], *cv_wroot[4];
  for (int i = 0; i < 4; ++i){
    cv_brel[i]  = (const float*)d_in[pi++];
    cv_wrel[i]  = (const float*)d_in[pi++];
    cv_wroot[i] = (const float*)d_in[pi++];
  }
  struct PoolP { const float *b_att,*b_lin,*le_b1,*le_w1,*le_w2,*le_w3,*w_att,*w_lin; } pp[2];
  for (int i = 0; i < 2; ++i){
    pp[i].b_att = (const float*)d_in[pi++];
    pp[i].b_lin = (const float*)d_in[pi++];
    pp[i].le_b1 = (const float*)d_in[pi++];
    pp[i].le_w1 = (const float*)d_in[pi++];
    pp[i].le_w2 = (const float*)d_in[pi++];
    pp[i].le_w3 = (const float*)d_in[pi++];
    pp[i].w_att = (const float*)d_in[pi++];
    pp[i].w_lin = (const float*)d_in[pi++];
  }
  const float* w1 = (const float*)d_in[pi++];
  const float* w2 = (const float*)d_in[pi++];
  const float* w3 = (const float*)d_in[pi++];
  const int* src = (const int*)d_in[n_in - 2];
  const int* dst = (const int*)d_in[n_in - 1];

  // bump allocator over d_ws
  char* wsp = (char*)d_ws;
  size_t off = 0;
  auto alloc = [&](long nf) -> float* {
    float* r = (float*)(wsp + off);
    off += (((size_t)nf * sizeof(float) + 255) & ~(size_t)255);
    return r;
  };
  float* agg  = alloc((long)NN * HF);
  float* h1   = alloc((long)NN * HF);
  float* h2   = alloc((long)NN * HF);
  float* degA = alloc(NN);
  float* degB = alloc(NN);
  float* xq   = alloc((long)NN * HF);
  float* xql  = alloc((long)NN * HF);
  float* aq   = alloc(NN);
  float* aj   = alloc(NN);
  float* ebuf = alloc(E2T);
  float* emax = alloc(NN);
  float* zbuf = alloc(NN);
  float* xc   = alloc((long)NN * HF);
  float* t1   = alloc(NN);
  float* t2   = alloc(NN);
  float* t3   = alloc(NN);
  float* fit  = alloc(NN);
  int*   idx1 = (int*)alloc(BGRAPH * KC1);
  float* S    = alloc((long)BGRAPH * NGN * NGN);
  float* Adn  = alloc((long)BGRAPH * NGN * NGN);
  float* Sp   = alloc((long)BGRAPH * NGN * KP1);
  float* Tb   = alloc((long)BGRAPH * NGN * KP1);
  float* A1   = alloc((long)BGRAPH * KP1 * KP1);
  float* xd1  = alloc((long)BGRAPH * KP1 * HF);
  float* aggd = alloc((long)BGRAPH * KP1 * HF);
  float* hd1  = alloc((long)BGRAPH * KP1 * HF);
  float* hd2  = alloc((long)BGRAPH * KP1 * HF);
  float* degD = alloc(BGRAPH * KP1);
  float* xqd  = alloc((long)BGRAPH * KP1 * HF);
  float* xqdl = alloc((long)BGRAPH * KP1 * HF);
  float* aqd  = alloc(BGRAPH * KP1);
  float* ajd  = alloc(BGRAPH * KP1);
  float* scod = alloc((long)BGRAPH * KP1 * KP1);
  float* xcd  = alloc((long)BGRAPH * KP1 * HF);
  float* t1d  = alloc(BGRAPH * KP1);
  float* t2d  = alloc(BGRAPH * KP1);
  float* t3d  = alloc(BGRAPH * KP1);
  float* fitd = alloc(BGRAPH * KP1);
  int*   idx2 = (int*)alloc(BGRAPH * KC2);
  float* Sp2  = alloc((long)BGRAPH * KP1 * KP2);
  float* T2b  = alloc((long)BGRAPH * KP1 * KP2);
  float* A2   = alloc((long)BGRAPH * KP2 * KP2);
  float* xd2  = alloc((long)BGRAPH * KP2 * HF);
  float* agg2 = alloc((long)BGRAPH * KP2 * HF);
  float* hd3  = alloc((long)BGRAPH * KP2 * HF);
  float* degE = alloc(BGRAPH * KP2);
  float* zcat = alloc(BGRAPH * 5 * HF);
  float* z1   = alloc(BGRAPH * HF);
  float* z2v  = alloc(BGRAPH * 64);

  // ---- in-degree (real edges only; clamp at use) ----
  k_fill<<<gdl(NN, TB), TB, 0, stream>>>(degA, 0.f, NN);
  k_deg<<<gdl(ETOT, TB), TB, 0, stream>>>(dst, degA, ETOT, 0);

  // ---- conv1: relu(mean-agg(x) @ w_rel + b + x @ w_root) ----
  k_fill<<<gdl((long)NN * HF, TB), TB, 0, stream>>>(agg, 0.f, (long)NN * HF);
  k_scatter_add<<<gdl((long)ETOT * HF, TB), TB, 0, stream>>>(x, src, dst, nullptr, agg, ETOT, 0, HF);
  k_scale_rows<<<gdl((long)NN * HF, TB), TB, 0, stream>>>(agg, degA, (long)NN * HF, HF);
  dim3 gN(NN / 16, HF / 16);
  k_wmma_gemm<<<gN, 32, 0, stream>>>(agg, HF, c1_wrel, x, HF, c1_wroot, c1_brel, h1, HF, HF, HF, 1);
  k_graph_sum<<<BGRAPH, HF, 0, stream>>>(h1, zcat, NGN, NGN, HF, 5 * HF, 0);

  // ---- convs[0] ----
  k_fill<<<gdl((long)NN * HF, TB), TB, 0, stream>>>(agg, 0.f, (long)NN * HF);
  k_scatter_add<<<gdl((long)ETOT * HF, TB), TB, 0, stream>>>(h1, src, dst, nullptr, agg, ETOT, 0, HF);
  k_scale_rows<<<gdl((long)NN * HF, TB), TB, 0, stream>>>(agg, degA, (long)NN * HF, HF);
  k_wmma_gemm<<<gN, 32, 0, stream>>>(agg, HF, cv_wrel[0], h1, HF, cv_wroot[0], cv_brel[0], h2, HF, HF, HF, 1);
  k_graph_sum<<<BGRAPH, HF, 0, stream>>>(h2, zcat, NGN, NGN, HF, 5 * HF, HF);

  // ---- ASAP pool (sparse) on h2 ----
  k_addc<<<gdl(NN, TB), TB, 0, stream>>>(degA, 1.f, degB, NN);  // +self loop
  k_fillu<<<gdl((long)NN * HF, TB), TB, 0, stream>>>((unsigned*)xq, 0u, (long)NN * HF);
  k_scatter_max<<<gdl((long)E2T * HF, TB), TB, 0, stream>>>(h2, src, dst, (unsigned*)xq, ETOT, NN, HF);
  k_decode<<<gdl((long)NN * HF, TB), TB, 0, stream>>>((unsigned*)xq, xq, (long)NN * HF);
  k_wmma_gemm<<<gN, 32, 0, stream>>>(xq, HF, pp[0].w_lin, nullptr, 0, nullptr, pp[0].b_lin, xql, HF, HF, HF, 0);
  k_matvec<<<gdl(NN, TB), TB, 0, stream>>>(xql, HF, pp[0].w_att, aq, NN, HF);
  k_matvec<<<gdl(NN, TB), TB, 0, stream>>>(h2, HF, pp[0].w_att + HF, aj, NN, HF);
  k_edge_e<<<gdl(E2T, TB), TB, 0, stream>>>(src, dst, aq, aj, pp[0].b_att, ebuf, ETOT, NN);
  k_fillu<<<gdl(NN, TB), TB, 0, stream>>>((unsigned*)emax, 0u, NN);
  k_edge_smax<<<gdl(E2T, TB), TB, 0, stream>>>(ebuf, dst, (unsigned*)emax, ETOT, NN);
  k_decode<<<gdl(NN, TB), TB, 0, stream>>>((unsigned*)emax, emax, NN);
  k_fill<<<gdl(NN, TB), TB, 0, stream>>>(zbuf, 0.f, NN);
  k_edge_exp<<<gdl(E2T, TB), TB, 0, stream>>>(ebuf, dst, emax, zbuf, ETOT, NN);
  k_edge_norm<<<gdl(E2T, TB), TB, 0, stream>>>(ebuf, dst, zbuf, ETOT, NN);
  k_fill<<<gdl((long)NN * HF, TB), TB, 0, stream>>>(xc, 0.f, (long)NN * HF);
  k_scatter_add<<<gdl((long)E2T * HF, TB), TB, 0, stream>>>(h2, src, dst, ebuf, xc, ETOT, NN, HF);
  k_matvec<<<gdl(NN, TB), TB, 0, stream>>>(xc, HF, pp[0].le_w1, t1, NN, HF);
  k_matvec<<<gdl(NN, TB), TB, 0, stream>>>(xc, HF, pp[0].le_w2, t2, NN, HF);
  k_matvec<<<gdl(NN, TB), TB, 0, stream>>>(xc, HF, pp[0].le_w3, t3, NN, HF);
  k_fit_init<<<gdl(NN, TB), TB, 0, stream>>>(t1, t2, degB, pp[0].le_b1, fit, NN);
  k_edge_sub<<<gdl(E2T, TB), TB, 0, stream>>>(t3, src, dst, fit, ETOT, NN);
  k_sigmoid<<<gdl(NN, TB), TB, 0, stream>>>(fit, NN);
  k_topk<<<BGRAPH, NGN, 0, stream>>>(fit, idx1, NGN, KC1);
  k_gather_x<<<gdl((long)BGRAPH * KP1 * HF, TB), TB, 0, stream>>>(xc, fit, idx1, xd1, NGN, KC1, KP1, HF);
  // coarsened adjacency A1 = Sp^T A Sp
  k_fill<<<gdl((long)BGRAPH * NGN * NGN, TB), TB, 0, stream>>>(S, 0.f, (long)BGRAPH * NGN * NGN);
  k_fill<<<gdl((long)BGRAPH * NGN * NGN, TB), TB, 0, stream>>>(Adn, 0.f, (long)BGRAPH * NGN * NGN);
  k_build_SA<<<gdl(E2T, TB), TB, 0, stream>>>(src, dst, ebuf, S, Adn, ETOT, NN);
  k_gather_Sp<<<gdl((long)BGRAPH * NGN * KP1, TB), TB, 0, stream>>>(S, idx1, Sp, NGN, NGN, KC1, KP1);
  dim3 gb1(NGN / 16, KP1 / 16, BGRAPH);
  k_wmma_bgemm<<<gb1, 32, 0, stream>>>(Adn, (long)NGN * NGN, NGN, 0, Sp, (long)NGN * KP1, KP1,
                                       Tb, (long)NGN * KP1, KP1, NGN);
  dim3 gb2(KP1 / 16, KP1 / 16, BGRAPH);
  k_wmma_bgemm<<<gb2, 32, 0, stream>>>(Sp, (long)NGN * KP1, KP1, 1, Tb, (long)NGN * KP1, KP1,
                                       A1, (long)KP1 * KP1, KP1, NGN);
  k_set_diag<<<gdl(BGRAPH * KC1, TB), TB, 0, stream>>>(A1, KP1, KC1);

  // ---- dense conv convs[1] ----
  k_dense_deg<<<gdl(BGRAPH * KP1, TB), TB, 0, stream>>>(A1, degD, KP1);
  dim3 gb3(KP1 / 16, HF / 16, BGRAPH);
  k_wmma_bgemm<<<gb3, 32, 0, stream>>>(A1, (long)KP1 * KP1, KP1, 1, xd1, (long)KP1 * HF, HF,
                                       aggd, (long)KP1 * HF, HF, KP1);
  k_scale_rows<<<gdl((long)BGRAPH * KP1 * HF, TB), TB, 0, stream>>>(aggd, degD, (long)BGRAPH * KP1 * HF, HF);
  dim3 gD1(BGRAPH * KP1 / 16, HF / 16);
  k_wmma_gemm<<<gD1, 32, 0, stream>>>(aggd, HF, cv_wrel[1], xd1, HF, cv_wroot[1], cv_brel[1], hd1, HF, HF, HF, 1);
  k_graph_sum<<<BGRAPH, HF, 0, stream>>>(hd1, zcat, KP1, KC1, HF, 5 * HF, 2 * HF);

  // ---- dense conv convs[2] ----
  k_wmma_bgemm<<<gb3, 32, 0, stream>>>(A1, (long)KP1 * KP1, KP1, 1, hd1, (long)KP1 * HF, HF,
                                       aggd, (long)KP1 * HF, HF, KP1);
  k_scale_rows<<<gdl((long)BGRAPH * KP1 * HF, TB), TB, 0, stream>>>(aggd, degD, (long)BGRAPH * KP1 * HF, HF);
  k_wmma_gemm<<<gD1, 32, 0, stream>>>(aggd, HF, cv_wrel[2], hd1, HF, cv_wroot[2], cv_brel[2], hd2, HF, HF, HF, 1);
  k_graph_sum<<<BGRAPH, HF, 0, stream>>>(hd2, zcat, KP1, KC1, HF, 5 * HF, 3 * HF);

  // ---- ASAP pool (dense) ----
  k_dense_xq<<<gdl((long)BGRAPH * KP1 * HF, TB), TB, 0, stream>>>(A1, hd2, xqd, KP1, HF);
  k_wmma_gemm<<<gD1, 32, 0, stream>>>(xqd, HF, pp[1].w_lin, nullptr, 0, nullptr, pp[1].b_lin, xqdl, HF, HF, HF, 0);
  k_matvec<<<gdl(BGRAPH * KP1, TB), TB, 0, stream>>>(xqdl, HF, pp[1].w_att, aqd, BGRAPH * KP1, HF);
  k_matvec<<<gdl(BGRAPH * KP1, TB), TB, 0, stream>>>(hd2, HF, pp[1].w_att + HF, ajd, BGRAPH * KP1, HF);
  k_dense_softmax<<<gdl(BGRAPH * KP1, TB), TB, 0, stream>>>(A1, aqd, ajd, pp[1].b_att, scod, KP1);
  k_wmma_bgemm<<<gb3, 32, 0, stream>>>(scod, (long)KP1 * KP1, KP1, 1, hd2, (long)KP1 * HF, HF,
                                       xcd, (long)KP1 * HF, HF, KP1);
  k_matvec<<<gdl(BGRAPH * KP1, TB), TB, 0, stream>>>(xcd, HF, pp[1].le_w1, t1d, BGRAPH * KP1, HF);
  k_matvec<<<gdl(BGRAPH * KP1, TB), TB, 0, stream>>>(xcd, HF, pp[1].le_w2, t2d, BGRAPH * KP1, HF);
  k_matvec<<<gdl(BGRAPH * KP1, TB), TB, 0, stream>>>(xcd, HF, pp[1].le_w3, t3d, BGRAPH * KP1, HF);
  k_dense_fit<<<gdl(BGRAPH * KP1, TB), TB, 0, stream>>>(t1d, t2d, t3d, A1, degD, pp[1].le_b1, fitd, KP1, KC1);
  k_topk<<<BGRAPH, KP1, 0, stream>>>(fitd, idx2, KP1, KC2);
  k_gather_x<<<gdl((long)BGRAPH * KP2 * HF, TB), TB, 0, stream>>>(xcd, fitd, idx2, xd2, KP1, KC2, KP2, HF);
  k_gather_Sp<<<gdl((long)BGRAPH * KP1 * KP2, TB), TB, 0, stream>>>(scod, idx2, Sp2, KP1, KP1, KC2, KP2);
  dim3 gb4(KP1 / 16, KP2 / 16, BGRAPH);
  k_wmma_bgemm<<<gb4, 32, 0, stream>>>(A1, (long)KP1 * KP1, KP1, 0, Sp2, (long)KP1 * KP2, KP2,
                                       T2b, (long)KP1 * KP2, KP2, KP1);
  dim3 gb5(KP2 / 16, KP2 / 16, BGRAPH);
  k_wmma_bgemm<<<gb5, 32, 0, stream>>>(Sp2, (long)KP1 * KP2, KP2, 1, T2b, (long)KP1 * KP2, KP2,
                                       A2, (long)KP2 * KP2, KP2, KP1);
  k_set_diag<<<gdl(BGRAPH * KC2, TB), TB, 0, stream>>>(A2, KP2, KC2);

  // ---- dense conv convs[3] ----
  k_dense_deg<<<gdl(BGRAPH * KP2, TB), TB, 0, stream>>>(A2, degE, KP2);
  dim3 gb6(KP2 / 16, HF / 16, BGRAPH);
  k_wmma_bgemm<<<gb6, 32, 0, stream>>>(A2, (long)KP2 * KP2, KP2, 1, xd2, (long)KP2 * HF, HF,
                                       agg2, (long)KP2 * HF, HF, KP2);
  k_scale_rows<<<gdl((long)BGRAPH * KP2 * HF, TB), TB, 0, stream>>>(agg2, degE, (long)BGRAPH * KP2 * HF, HF);
  dim3 gD2(BGRAPH * KP2 / 16, HF / 16);
  k_wmma_gemm<<<gD2, 32, 0, stream>>>(agg2, HF, cv_wrel[3], xd2, HF, cv_wroot[3], cv_brel[3], hd3, HF, HF, HF, 1);
  k_graph_sum<<<BGRAPH, HF, 0, stream>>>(hd3, zcat, KP2, KC2, HF, 5 * HF, 4 * HF);

  // ---- MLP head ----
  dim3 gm1(BGRAPH / 16, HF / 16);
  k_wmma_gemm<<<gm1, 32, 0, stream>>>(zcat, 5 * HF, w1, nullptr, 0, nullptr, b1, z1, HF, HF, 5 * HF, 1);
  dim3 gm2(BGRAPH / 16, 64 / 16);
  k_wmma_gemm<<<gm2, 32, 0, stream>>>(z1, HF, w2, nullptr, 0, nullptr, b2, z2v, 64, 64, HF, 1);
  k_head<<<1, 32, 0, stream>>>(z2v, w3, b3, (float*)d_out);
}